// Attention_46858093199643
// MI455X (gfx1250) — compile-verified
//
#include <hip/hip_runtime.h>
#include <hip/hip_bf16.h>

typedef __attribute__((ext_vector_type(2))) float v2f;
typedef __attribute__((ext_vector_type(8))) float v8f;

#define H_DIM 128
#define A_DIM 256
#define NS 1024
#define NE 512
#define M_DIM 1024

// ---------------------------------------------------------------------------
// Hardware transcendental tanh (gfx1250 V_TANH_F32).
// TRANS-op hazard rule (CDNA5 ISA 7.4): 1 independent op or V_NOP after a
// trans op before its output is consumed / sources overwritten. The compiler
// cannot see through inline asm, so the V_NOP is embedded here.
// ---------------------------------------------------------------------------
__device__ __forceinline__ float hw_tanh(float x) {
    float y;
    asm volatile("v_tanh_f32 %0, %1\n\tv_nop" : "=v"(y) : "v"(x));
    return y;
}

// ---------------------------------------------------------------------------
// Y[n, h] = sum_k X[n,k] * W[h, koff+k]  (+ bias[h])     (X: rows x K, W row-major ldw)
// One wave (32 lanes) computes one 16x16 tile via V_WMMA_F32_16X16X4_F32.
// A frag (16x4 f32): lane -> M = lane&15, VGPR j -> K = k + j + 2*(lane>>4)
// B frag (4x16 f32): lane -> N = lane&15, VGPR j -> K = k + j + 2*(lane>>4)
// C/D (16x16 f32):  VGPR r -> M = r + 8*(lane>>4), N = lane&15
// ---------------------------------------------------------------------------
__global__ void k_gemm_xwT(const float* __restrict__ X, int ldx,
                           const float* __restrict__ W, int ldw, int koff,
                           const float* __restrict__ bias,
                           float* __restrict__ Y, int ldy,
                           int tilesN, int totalTiles, int K)
{
    int tile = blockIdx.x * 4 + (threadIdx.x >> 5);
    if (tile >= totalTiles) return;           // whole-wave uniform: EXEC stays all-1s
    int lane = threadIdx.x & 31;
    int g    = lane >> 4;
    int l15  = lane & 15;
    int tm = tile / tilesN, tn = tile % tilesN;

    const float* arow = X + (size_t)(tm * 16 + l15) * ldx;
    const float* brow = W + (size_t)(tn * 16 + l15) * ldw + koff;

    v8f acc = {};
    for (int k = 0; k < K; k += 4) {
        int kk = k + 2 * g;
        v2f a = *(const v2f*)(arow + kk);
        v2f b = *(const v2f*)(brow + kk);
        acc = __builtin_amdgcn_wmma_f32_16x16x4_f32(false, a, false, b,
                                                    (short)0, acc, false, false);
    }

    int col = tn * 16 + l15;
    float bv = bias ? bias[col] : 0.0f;
    #pragma unroll
    for (int r = 0; r < 8; ++r) {
        int row = tm * 16 + r + 8 * g;
        Y[(size_t)row * ldy + col] = acc[r] + bv;
    }
}

// ---------------------------------------------------------------------------
// st[m, n] = sum_h v[h] * tanh(a[n,h] + b[m,h]) + c      (transposed score store)
// 16x16 (n,m) tile per 256-thread block; a/b tiles + v staged in LDS (padded).
// Dominant cost of the whole problem: ~200M hardware tanh evaluations.
// ---------------------------------------------------------------------------
__global__ void k_score(const float* __restrict__ a,   // N x 128
                        const float* __restrict__ b,   // M x 128
                        const float* __restrict__ v,
                        const float* __restrict__ cPtr,
                        float* __restrict__ st, int N)
{
    __shared__ float as_[16][132];  // pad to break 64-bank conflicts
    __shared__ float bs_[16][132];
    __shared__ float vs_[128];

    int tid = threadIdx.x;
    int n0 = blockIdx.x * 16;
    int m0 = blockIdx.y * 16;

    for (int i = tid; i < 16 * 128; i += 256) {
        int r = i >> 7, h = i & 127;
        as_[r][h] = a[(size_t)(n0 + r) * H_DIM + h];
        bs_[r][h] = b[(size_t)(m0 + r) * H_DIM + h];
    }
    if (tid < 128) vs_[tid] = v[tid];
    __syncthreads();

    int tn = tid & 15;   // n within tile (contiguous store)
    int tmr = tid >> 4;  // m within tile
    float s = 0.0f;
    #pragma unroll 4
    for (int h = 0; h < 128; ++h)
        s += vs_[h] * hw_tanh(as_[tn][h] + bs_[tmr][h]);

    st[(size_t)(m0 + tmr) * N + (n0 + tn)] = s + cPtr[0];
}

// ---------------------------------------------------------------------------
// Row-wise softmax over contiguous rows of length N (one block per row).
// ---------------------------------------------------------------------------
__global__ void k_softmax_row(float* __restrict__ st, int N)
{
    __shared__ float red[256];
    float* p = st + (size_t)blockIdx.x * N;
    int tid = threadIdx.x;

    float mx = -3.0e38f;
    for (int i = tid; i < N; i += 256) mx = fmaxf(mx, p[i]);
    red[tid] = mx; __syncthreads();
    for (int s = 128; s > 0; s >>= 1) {
        if (tid < s) red[tid] = fmaxf(red[tid], red[tid + s]);
        __syncthreads();
    }
    mx = red[0]; __syncthreads();

    float sum = 0.0f;
    for (int i = tid; i < N; i += 256) sum += __expf(p[i] - mx);
    red[tid] = sum; __syncthreads();
    for (int s = 128; s > 0; s >>= 1) {
        if (tid < s) red[tid] += red[tid + s];
        __syncthreads();
    }
    float inv = 1.0f / red[0];

    for (int i = tid; i < N; i += 256) p[i] = __expf(p[i] - mx) * inv;
}

// ---------------------------------------------------------------------------
// Y[m, h] = sum_k Aw[m,k] * B[k,h]       (Aw: M x K row-major, B: K x 128)
// ---------------------------------------------------------------------------
__global__ void k_gemm_ab(const float* __restrict__ Aw, int lda,
                          const float* __restrict__ B,
                          float* __restrict__ Y, int K, int totalTiles)
{
    int tile = blockIdx.x * 4 + (threadIdx.x >> 5);
    if (tile >= totalTiles) return;
    int lane = threadIdx.x & 31;
    int g = lane >> 4, l15 = lane & 15;
    int tm = tile >> 3, tn = tile & 7;        // tilesN = 128/16 = 8

    const float* arow = Aw + (size_t)(tm * 16 + l15) * lda;
    int n = tn * 16 + l15;

    v8f acc = {};
    for (int k = 0; k < K; k += 4) {
        int kk = k + 2 * g;
        v2f a = *(const v2f*)(arow + kk);
        v2f b;
        b.x = B[(size_t)kk * H_DIM + n];
        b.y = B[(size_t)(kk + 1) * H_DIM + n];
        acc = __builtin_amdgcn_wmma_f32_16x16x4_f32(false, a, false, b,
                                                    (short)0, acc, false, false);
    }
    #pragma unroll
    for (int r = 0; r < 8; ++r)
        Y[(size_t)(tm * 16 + r + 8 * g) * H_DIM + n] = acc[r];
}

// ---------------------------------------------------------------------------
// out[m, c] = tanh( sum_j cat[m,j] * W_lin[c,j] + b_lin[c] ),
// cat = [attender | ctx_s | ctx_e], K = 384, out: 1024 x 256
// ---------------------------------------------------------------------------
__global__ void k_final(const float* __restrict__ att,
                        const float* __restrict__ cs,
                        const float* __restrict__ ce,
                        const float* __restrict__ Wl,
                        const float* __restrict__ bl,
                        float* __restrict__ out)
{
    int tile = blockIdx.x * 4 + (threadIdx.x >> 5);  // 1024 tiles: 64 x 16
    if (tile >= 1024) return;
    int lane = threadIdx.x & 31;
    int g = lane >> 4, l15 = lane & 15;
    int tm = tile >> 4, tn = tile & 15;

    int mrow = tm * 16 + l15;
    const float* wrow = Wl + (size_t)(tn * 16 + l15) * (3 * H_DIM);

    v8f acc = {};
    for (int k = 0; k < 3 * H_DIM; k += 4) {
        int seg = k >> 7;                      // uniform: k%4==0, so (k+2g)>>7 == k>>7
        const float* base = (seg == 0) ? att : (seg == 1) ? cs : ce;
        int kk = k + 2 * g;
        v2f a = *(const v2f*)(base + (size_t)mrow * H_DIM + (kk & 127));
        v2f b = *(const v2f*)(wrow + kk);
        acc = __builtin_amdgcn_wmma_f32_16x16x4_f32(false, a, false, b,
                                                    (short)0, acc, false, false);
    }

    int col = tn * 16 + l15;
    float bv = bl[col];
    #pragma unroll
    for (int r = 0; r < 8; ++r) {
        int row = tm * 16 + r + 8 * g;
        out[(size_t)row * A_DIM + col] = hw_tanh(acc[r] + bv);
    }
}

// ---------------------------------------------------------------------------
extern "C" void kernel_launch(void* const* d_in, const int* in_sizes, int n_in,
                              void* d_out, int out_size, void* d_ws, size_t ws_size,
                              hipStream_t stream)
{
    (void)in_sizes; (void)n_in; (void)out_size; (void)ws_size;
    const float* stmts = (const float*)d_in[0];   // (1024, 128)
    const float* eres  = (const float*)d_in[1];   // (512, 128)
    const float* att   = (const float*)d_in[2];   // (1024, 128)
    const float* Ws    = (const float*)d_in[3];   // (128, 256)
    const float* bs    = (const float*)d_in[4];   // (128,)
    const float* vs    = (const float*)d_in[5];   // (128,)
    const float* csc   = (const float*)d_in[6];   // scalar
    const float* We    = (const float*)d_in[7];   // (128, 256)
    const float* be    = (const float*)d_in[8];   // (128,)
    const float* ve    = (const float*)d_in[9];   // (128,)
    const float* cec   = (const float*)d_in[10];  // scalar
    const float* Wl    = (const float*)d_in[11];  // (256, 384)
    const float* bl    = (const float*)d_in[12];  // (256,)
    float* out = (float*)d_out;                   // (1024, 256)

    float* ws    = (float*)d_ws;
    float* a_s   = ws;                              // NS x H
    float* a_e   = a_s  + (size_t)NS * H_DIM;       // NE x H
    float* b_s   = a_e  + (size_t)NE * H_DIM;       // M x H
    float* b_e   = b_s  + (size_t)M_DIM * H_DIM;    // M x H
    float* st_ss = b_e  + (size_t)M_DIM * H_DIM;    // M x NS (transposed scores)
    float* st_es = st_ss + (size_t)M_DIM * NS;      // M x NE
    float* ctx_s = st_es + (size_t)M_DIM * NE;      // M x H
    float* ctx_e = ctx_s + (size_t)M_DIM * H_DIM;   // M x H

    // a = x1 @ W1^T ; b = attender @ W2^T + bias   (W1 = Wc[:, :H], W2 = Wc[:, H:])
    k_gemm_xwT<<<128, 128, 0, stream>>>(stmts, H_DIM, Ws, 2 * H_DIM, 0,     nullptr, a_s, H_DIM, 8, 512, H_DIM);
    k_gemm_xwT<<< 64, 128, 0, stream>>>(eres,  H_DIM, We, 2 * H_DIM, 0,     nullptr, a_e, H_DIM, 8, 256, H_DIM);
    k_gemm_xwT<<<128, 128, 0, stream>>>(att,   H_DIM, Ws, 2 * H_DIM, H_DIM, bs,      b_s, H_DIM, 8, 512, H_DIM);
    k_gemm_xwT<<<128, 128, 0, stream>>>(att,   H_DIM, We, 2 * H_DIM, H_DIM, be,      b_e, H_DIM, 8, 512, H_DIM);

    // scores (transposed: st[m, n])
    dim3 gs1(NS / 16, M_DIM / 16);
    k_score<<<gs1, 256, 0, stream>>>(a_s, b_s, vs, csc, st_ss, NS);
    dim3 gs2(NE / 16, M_DIM / 16);
    k_score<<<gs2, 256, 0, stream>>>(a_e, b_e, ve, cec, st_es, NE);

    // softmax over attendee axis (= contiguous rows of st)
    k_softmax_row<<<M_DIM, 256, 0, stream>>>(st_ss, NS);
    k_softmax_row<<<M_DIM, 256, 0, stream>>>(st_es, NE);

    // ctx[m, h] = sum_n w[m, n] * attendee[n, h]
    k_gemm_ab<<<128, 128, 0, stream>>>(st_ss, NS, stmts, ctx_s, NS, 512);
    k_gemm_ab<<<128, 128, 0, stream>>>(st_es, NE, eres,  ctx_e, NE, 512);

    // out = tanh([att | ctx_s | ctx_e] @ W_lin^T + b_lin)
    k_final<<<256, 128, 0, stream>>>(att, ctx_s, ctx_e, Wl, bl, out);
}